// DifferentiableForwardModel_12816182411700
// MI455X (gfx1250) — compile-verified
//
#include <hip/hip_runtime.h>

typedef __attribute__((ext_vector_type(2))) float v2f;
typedef __attribute__((ext_vector_type(8))) float v8f;

typedef __attribute__((address_space(1))) int as1_int;  // global (HIP __device__) int
typedef __attribute__((address_space(3))) int as3_int;  // LDS (HIP __shared__) int

#define HH   256
#define WW   256
#define NLAM 96
#define NB   4
#define HO   288
#define WO   288
#define KS   15
#define KR   7
#define TILES_X 18              // 288/16
#define TILES_PER_CH (18*18)    // 324
#define NCH 16                  // B * 4 k-sequences

#if defined(__has_builtin)
#  if __has_builtin(__builtin_amdgcn_global_load_async_to_lds_b32) && \
      __has_builtin(__builtin_amdgcn_s_wait_asynccnt)
#    define USE_ASYNC_LDS 1
#  endif
#endif
#ifndef USE_ASYNC_LDS
#  define USE_ASYNC_LDS 0
#endif

__device__ __forceinline__ float tap(const float* __restrict__ im, int yy, int xx) {
  return (yy >= 0 && yy < HH && xx >= 0 && xx < WW) ? __ldg(im + yy * WW + xx) : 0.f;
}

// ---------------------------------------------------------------------------
// Kernel 1: fused pad + bilinear shift + sum over lambda, ALL 4 k-sequences
// per thread so each cube tile is read once (16 taps/lambda share cache lines).
// acc[ch][y][x], ch = b*4 + k.
// ---------------------------------------------------------------------------
__global__ void __launch_bounds__(256)
shift_accum_kernel(const float* __restrict__ cube,
                   const float* __restrict__ dx,
                   const float* __restrict__ dy,
                   float* __restrict__ acc) {
  __shared__ float sdx[4 * NLAM];
  __shared__ float sdy[4 * NLAM];
  const int b = blockIdx.z;
  const int x = blockIdx.x * 16 + (threadIdx.x & 15);
  const int y = blockIdx.y * 16 + (threadIdx.x >> 4);

  for (int i = threadIdx.x; i < 4 * NLAM; i += blockDim.x) {
    sdx[i] = dx[i];           // dx,dy are (4, NLAM) flat: k*NLAM + l
    sdy[i] = dy[i];
  }
  __syncthreads();

  const float* __restrict__ img = cube + (size_t)b * NLAM * HH * WW;
  // clamped coords for prefetching the next-lambda image
  int py = y - 16; py = py < 0 ? 0 : (py > HH - 1 ? HH - 1 : py);
  int px = x - 16; px = px < 0 ? 0 : (px > WW - 1 ? WW - 1 : px);

  float a[4] = {0.f, 0.f, 0.f, 0.f};
  const float fy = (float)(y - 16);
  const float fx = (float)(x - 16);

  for (int l = 0; l < NLAM; ++l) {
    const float* __restrict__ im = img + (size_t)l * HH * WW;
    if (l + 1 < NLAM) {
      __builtin_prefetch(im + HH * WW + py * WW + px, 0, 1); // global_prefetch_b8
    }
    #pragma unroll
    for (int k = 0; k < 4; ++k) {
      const float sy = fy - sdy[k * NLAM + l];
      const float sx = fx - sdx[k * NLAM + l];
      const float fy0 = floorf(sy), fx0 = floorf(sx);
      const int   y0 = (int)fy0,    x0 = (int)fx0;
      const float wy = sy - fy0,    wx = sx - fx0;
      const float p00 = tap(im, y0,     x0);
      const float p01 = tap(im, y0,     x0 + 1);
      const float p10 = tap(im, y0 + 1, x0);
      const float p11 = tap(im, y0 + 1, x0 + 1);
      const float top = p00 + wx * (p01 - p00);
      const float bot = p10 + wx * (p11 - p10);
      a[k] += top + wy * (bot - top);
    }
  }

  #pragma unroll
  for (int k = 0; k < 4; ++k) {
    acc[(size_t)(b * 4 + k) * HO * WO + (size_t)y * WO + x] = a[k];
  }
}

// ---------------------------------------------------------------------------
// Kernel 2: 15x15 PSF correlation via V_WMMA_F32_16X16X4_F32.
// One wave per 16x16 output tile. Per PSF column j:
//   out_tile(16x16) += A_j(16x32) x B_j(32x16)
// A_j[m,r] = psf[r-m, j] (banded Toeplitz), B_j[r,n] = patch[r, n+j],
// patch = 32x32 LDS slab of acc rows [y0-7, y0+24], cols [x0-7, x0+24].
// Interior tiles stage the patch with GLOBAL_LOAD_ASYNC_TO_LDS_B32.
// ---------------------------------------------------------------------------
__global__ void __launch_bounds__(256)
psf_conv_wmma_kernel(const float* __restrict__ acc,
                     const float* __restrict__ psf,
                     float* __restrict__ out) {
  __shared__ float spsf[KS * KS];
  __shared__ float patch[8][32 * 33];   // +1 pad to dodge bank conflicts

  const int lane = threadIdx.x & 31;
  const int wv   = threadIdx.x >> 5;

  for (int i = threadIdx.x; i < KS * KS; i += blockDim.x) spsf[i] = psf[i];

  const int tile = blockIdx.x * 8 + wv;           // 648 blocks * 8 waves = 5184
  const int ch   = tile / TILES_PER_CH;
  const int t    = tile - ch * TILES_PER_CH;
  const int y0   = (t / TILES_X) * 16;
  const int x0   = (t % TILES_X) * 16;

  const float* __restrict__ src = acc + (size_t)ch * HO * WO;
  float* __restrict__ ld = patch[wv];

  // stage 32x32 patch: rows y0-7..y0+24, cols x0-7..x0+24 (zero-filled OOB)
  const bool interior = (y0 >= KR) && (y0 + 32 - KR <= HO) &&
                        (x0 >= KR) && (x0 + 32 - KR <= WO);
#if USE_ASYNC_LDS
  if (interior) {
    const float* g0 = src + (size_t)(y0 - KR) * WO + (x0 - KR + lane);
    #pragma unroll 4
    for (int r = 0; r < 32; ++r) {
      __builtin_amdgcn_global_load_async_to_lds_b32(
          (as1_int*)(uintptr_t)(g0 + (size_t)r * WO),
          (as3_int*)(uintptr_t)&ld[r * 33 + lane],
          0, 0);
    }
    __builtin_amdgcn_s_wait_asynccnt(0);
  } else
#else
  (void)interior;
#endif
  {
    #pragma unroll 4
    for (int r = 0; r < 32; ++r) {
      const int yy = y0 - KR + r;
      const int xx = x0 - KR + lane;
      float v = 0.f;
      if (yy >= 0 && yy < HO && xx >= 0 && xx < WO) v = src[(size_t)yy * WO + xx];
      ld[r * 33 + lane] = v;
    }
  }
  __syncthreads();

  const int m     = lane & 15;          // A row / B col / D col
  const int khalf = (lane >> 4) << 1;   // K lane-half offset: 0 or 2

  v8f c = {0.f, 0.f, 0.f, 0.f, 0.f, 0.f, 0.f, 0.f};

  for (int j = 0; j < KS; ++j) {
    #pragma unroll
    for (int kc = 0; kc < 8; ++kc) {
      const int r0 = 4 * kc + khalf;    // K index for frag element 0
      const int r1 = r0 + 1;            // K index for frag element 1
      const int i0 = r0 - m;
      const int i1 = r1 - m;
      v2f a, bb;
      a.x = (i0 >= 0 && i0 < KS) ? spsf[i0 * KS + j] : 0.f;
      a.y = (i1 >= 0 && i1 < KS) ? spsf[i1 * KS + j] : 0.f;
      bb.x = ld[r0 * 33 + m + j];
      bb.y = ld[r1 * 33 + m + j];
      c = __builtin_amdgcn_wmma_f32_16x16x4_f32(
              /*neg_a=*/false, a, /*neg_b=*/false, bb,
              /*c_mod=*/(short)0, c, /*reuse_a=*/false, /*reuse_b=*/false);
    }
  }

  // D layout: VGPR p -> row (p | p+8 by lane half), col = lane&15
  const int row_base = (lane >> 4) * 8;
  float* __restrict__ dst = out + (size_t)ch * HO * WO;
  #pragma unroll
  for (int p = 0; p < 8; ++p) {
    dst[(size_t)(y0 + row_base + p) * WO + (x0 + m)] = c[p];
  }
}

// ---------------------------------------------------------------------------
extern "C" void kernel_launch(void* const* d_in, const int* in_sizes, int n_in,
                              void* d_out, int out_size, void* d_ws, size_t ws_size,
                              hipStream_t stream) {
  (void)in_sizes; (void)n_in; (void)out_size; (void)ws_size;
  const float* cube = (const float*)d_in[0];   // (4, 96, 256, 256)
  const float* dx   = (const float*)d_in[1];   // (4, 96)
  const float* dy   = (const float*)d_in[2];   // (4, 96)
  const float* psf  = (const float*)d_in[3];   // (15, 15)
  float* out = (float*)d_out;                  // (4, 4, 288, 288)
  float* acc = (float*)d_ws;                   // 16*288*288 floats = 5.3 MB scratch

  dim3 g1(TILES_X, TILES_X, NB);
  shift_accum_kernel<<<g1, 256, 0, stream>>>(cube, dx, dy, acc);

  const int nblocks = (NCH * TILES_PER_CH) / 8; // 648
  psf_conv_wmma_kernel<<<nblocks, 256, 0, stream>>>(acc, psf, out);
}